// PsuedoConvFace_79757542686874
// MI455X (gfx1250) — compile-verified
//
#include <hip/hip_runtime.h>

#define B_       8
#define C_IN     128
#define C_OUT    256
#define F_FULL   16384
#define F_POOL   8192
#define K_NEI    32
#define BN_EPS   1e-5f

typedef __attribute__((ext_vector_type(2))) float v2f;
typedef __attribute__((ext_vector_type(8))) float v8f;

// -------- Kernel 1: transpose fea [B, C, F] -> feaT [B, F, C] --------
// Makes each gathered "column" a contiguous 512B row.
__global__ void transpose_fea(const float* __restrict__ fea,
                              float* __restrict__ feaT) {
    __shared__ float tile[32][33];
    const int fb = blockIdx.x * 32;
    const int cb = blockIdx.y * 32;
    const int b  = blockIdx.z;
    const int tx = threadIdx.x, ty = threadIdx.y;
#pragma unroll
    for (int j = 0; j < 4; ++j) {
        int c = cb + ty + j * 8;
        tile[ty + j * 8][tx] = fea[((size_t)b * C_IN + c) * F_FULL + fb + tx];
    }
    __syncthreads();
#pragma unroll
    for (int j = 0; j < 4; ++j) {
        int f = fb + ty + j * 8;
        feaT[((size_t)b * F_FULL + f) * C_IN + cb + tx] = tile[tx][ty + j * 8];
    }
}

// -------- Kernel 2: gather + neighbor-sum --------
// One wave32 per (b, f). Lane k holds index ring_n[b,f,k]; rows broadcast
// via __shfl. Each lane accumulates a float4 slice of the 128-float row.
__global__ void gather_sum(const float* __restrict__ feaT,
                           const int*   __restrict__ ring,
                           const int*   __restrict__ pool_idx,
                           float*       __restrict__ summed) {
    const int wid  = (blockIdx.x * blockDim.x + threadIdx.x) >> 5; // b*F_POOL+f
    const int lane = threadIdx.x & 31;
    const int b    = wid / F_POOL;
    const int f    = wid - b * F_POOL;

    const int myIdx = ring[(size_t)wid * K_NEI + lane];
    const float4* base = (const float4*)(feaT + (size_t)b * F_FULL * C_IN);

    const int pc = pool_idx[f];
    float4 acc = base[(size_t)pc * (C_IN / 4) + lane];

#pragma unroll 4
    for (int k = 0; k < K_NEI; ++k) {
        int r = __shfl(myIdx, k, 32);
        if (k + 1 < K_NEI) {
            int rn = __shfl(myIdx, k + 1, 32);
            __builtin_prefetch((const void*)(base + (size_t)rn * (C_IN / 4) + lane), 0, 0);
        }
        float4 v = base[(size_t)r * (C_IN / 4) + lane];
        acc.x += v.x; acc.y += v.y; acc.z += v.z; acc.w += v.w;
    }
    float4* out = (float4*)(summed + (size_t)wid * C_IN);
    out[lane] = acc;
}

// -------- Kernel 3: WMMA f32 GEMM + fused BatchNorm + ReLU --------
// Per batch: D[o,f] = sum_c W[o,c] * summed[b,f,c]. One wave per 16x16 tile,
// 4 waves/block (M=16, N=64 per block). A and per-wave B tiles staged in LDS
// with stride 132 (pad) to avoid bank conflicts.
#define LDA 132
__global__ void gemm_bn_relu(const float* __restrict__ Wm,
                             const float* __restrict__ summed,
                             const float* __restrict__ bias,
                             const float* __restrict__ gamma,
                             const float* __restrict__ beta,
                             const float* __restrict__ mean,
                             const float* __restrict__ var,
                             float* __restrict__ out) {
    __shared__ float ldsA[16 * LDA];
    __shared__ float ldsB[4][16 * LDA];

    const int tid   = threadIdx.x;
    const int w     = tid >> 5;
    const int lane  = tid & 31;
    const int hi    = lane >> 4;   // 0: lanes 0-15, 1: lanes 16-31
    const int l15   = lane & 15;
    const int b     = blockIdx.z;
    const int Mbase = blockIdx.y * 16;
    const int Nwave = blockIdx.x * 64 + w * 16;

    // Stage A tile: W[Mbase..Mbase+15][0..127]  (coalesced float4 loads)
    for (int e = tid; e < 16 * 32; e += 128) {
        int row = e >> 5;
        int c4  = (e & 31) * 4;
        const float* src = Wm + (size_t)(Mbase + row) * C_IN + c4;
        float* dst = &ldsA[row * LDA + c4];
        dst[0] = src[0]; dst[1] = src[1]; dst[2] = src[2]; dst[3] = src[3];
    }
    // Stage per-wave B tile: summed[b, Nwave..Nwave+15, 0..127]
    for (int r = 0; r < 16; ++r) {
        const float* src = summed + ((size_t)b * F_POOL + Nwave + r) * C_IN + lane * 4;
        float* dst = &ldsB[w][r * LDA + lane * 4];
        dst[0] = src[0]; dst[1] = src[1]; dst[2] = src[2]; dst[3] = src[3];
    }
    __syncthreads();

    // 32 x V_WMMA_F32_16X16X4_F32 over K = C_IN = 128.
    // ISA f32 A(16x4)/B(4x16) layout: lanes 0-15 hold K pair {k,k+1},
    // lanes 16-31 hold K pair {k+2,k+3}; M/N = lane & 15.
    v8f acc = {};
#pragma unroll
    for (int k = 0; k < C_IN; k += 4) {
        const float* Ar = &ldsA[l15 * LDA + k + 2 * hi];
        const float* Br = &ldsB[w][l15 * LDA + k + 2 * hi];
        v2f a;  a.x  = Ar[0]; a.y  = Ar[1];
        v2f bb; bb.x = Br[0]; bb.y = Br[1];
        acc = __builtin_amdgcn_wmma_f32_16x16x4_f32(
            /*neg_a=*/false, a, /*neg_b=*/false, bb,
            /*c_mod=*/(short)0, acc, /*reuse_a=*/false, /*reuse_b=*/false);
    }

    // Epilogue: fold conv-bias + BN (inference) + ReLU.
    // C/D layout: VGPR r, lane -> row o = Mbase + r + 8*hi, col f = Nbase + l15.
    const int f = Nwave + l15;
#pragma unroll
    for (int r = 0; r < 8; ++r) {
        int o = Mbase + r + 8 * hi;
        float s  = gamma[o] * rsqrtf(var[o] + BN_EPS);
        float sh = (bias[o] - mean[o]) * s + beta[o];
        float v  = acc[r] * s + sh;
        out[((size_t)b * C_OUT + o) * F_POOL + f] = fmaxf(v, 0.0f);
    }
}

extern "C" void kernel_launch(void* const* d_in, const int* in_sizes, int n_in,
                              void* d_out, int out_size, void* d_ws, size_t ws_size,
                              hipStream_t stream) {
    const float* fea   = (const float*)d_in[0];
    const int*   ring  = (const int*)  d_in[1];
    const int*   pidx  = (const int*)  d_in[2];
    const float* Wm    = (const float*)d_in[3];
    const float* bias  = (const float*)d_in[4];
    const float* gamma = (const float*)d_in[5];
    const float* beta  = (const float*)d_in[6];
    const float* mean  = (const float*)d_in[7];
    const float* var   = (const float*)d_in[8];
    float* out = (float*)d_out;

    // Workspace layout: feaT (64 MB) | summed (32 MB)
    float* feaT   = (float*)d_ws;
    float* summed = feaT + (size_t)B_ * F_FULL * C_IN;

    transpose_fea<<<dim3(F_FULL / 32, C_IN / 32, B_), dim3(32, 8), 0, stream>>>(fea, feaT);
    gather_sum<<<dim3((B_ * F_POOL) / 8), 256, 0, stream>>>(feaT, ring, pidx, summed);
    gemm_bn_relu<<<dim3(F_POOL / 64, C_OUT / 16, B_), 128, 0, stream>>>(
        Wm, summed, bias, gamma, beta, mean, var, out);
}